// Dictionary_41807211659920
// MI455X (gfx1250) — compile-verified
//
#include <hip/hip_runtime.h>
#include <math.h>

typedef __attribute__((ext_vector_type(2))) float v2f;
typedef __attribute__((ext_vector_type(4))) float v4f;
typedef __attribute__((ext_vector_type(8))) float v8f;

#define NUM_TASKS 1000
#define NFEAT 256
#define HID 128
#define BATCH 4096
#define WAVES_PER_BLOCK 4
#define NTILES 8  // 128 hidden cols / 16 per WMMA tile

__global__ __launch_bounds__(WAVES_PER_BLOCK * 32)
void mtl_fused_kernel(const float* __restrict__ x,
                      const int* __restrict__ task_ids,
                      const float* __restrict__ l1_emb,
                      const float* __restrict__ l1_bias,
                      const float* __restrict__ l2_emb,
                      const float* __restrict__ l2_bias,
                      float* __restrict__ out) {
    __shared__ float xs[WAVES_PER_BLOCK][NFEAT];

    const int wave = threadIdx.x >> 5;
    const int lane = threadIdx.x & 31;
    const int sample = blockIdx.x * WAVES_PER_BLOCK + wave;   // grid sized exactly

    const int task = task_ids[sample];
    const float* __restrict__ xp = x + (size_t)sample * NFEAT;

    // Stage this wave's x[256] into LDS with two b128 loads per lane.
    {
        v4f a = *(const v4f*)(xp + lane * 4);
        v4f b = *(const v4f*)(xp + 128 + lane * 4);
        *(v4f*)(&xs[wave][lane * 4]) = a;
        *(v4f*)(&xs[wave][128 + lane * 4]) = b;
    }
    __syncthreads();

    const float* __restrict__ w1 = l1_emb + (size_t)task * (NFEAT * HID);

    // 8 accumulator tiles: h columns [16t, 16t+16)
    v8f acc[NTILES];
    v8f zero = {};
#pragma unroll
    for (int t = 0; t < NTILES; ++t) acc[t] = zero;

    const int khalf = (lane >> 4) * 2;  // lanes 0-15 -> K0/K1 ; lanes 16-31 -> K2/K3
    const int nl = lane & 15;           // N index within a 16-wide tile

    for (int k0 = 0; k0 < NFEAT; k0 += 4) {
        // A (16x4 f32): broadcast the sample's x chunk into all 16 M-rows.
        // v0 = x[k0 + khalf], v1 = x[k0 + khalf + 1]
        v2f A;
        A.x = xs[wave][k0 + khalf];
        A.y = xs[wave][k0 + khalf + 1];

        // B tile base for this K-step: row (k0+khalf), column nl
        const float* __restrict__ wk = w1 + (size_t)(k0 + khalf) * HID + nl;

        // Prefetch next K-chunk of w1 (lowers to global_prefetch_b8)
        if (k0 + 4 < NFEAT) {
            __builtin_prefetch(wk + 4 * HID, 0, 0);
        }

#pragma unroll
        for (int t = 0; t < NTILES; ++t) {
            // B (4x16 f32): v0 = w1[k0+khalf, 16t+nl], v1 = w1[k0+khalf+1, 16t+nl]
            v2f B;
            B.x = wk[t * 16];
            B.y = wk[t * 16 + HID];
            // 8 args: (neg_a, A, neg_b, B, c_mod, C, reuse_a, reuse_b)
            acc[t] = __builtin_amdgcn_wmma_f32_16x16x4_f32(
                false, A, false, B, (short)0, acc[t], false, false);
        }
    }

    // Every D row equals h (all A rows were the sample), so acc[t][0] in any
    // lane L is h[16t + (L&15)]. Layer 2: bias, exact-erf GELU, dot with w2.
    const float* __restrict__ b1 = l1_bias + (size_t)task * HID;
    const float* __restrict__ w2 = l2_emb + (size_t)task * HID;

    float s = 0.0f;
#pragma unroll
    for (int t = 0; t < NTILES; ++t) {
        const int n = t * 16 + nl;
        float h = acc[t][0] + b1[n];
        float g = 0.5f * h * (1.0f + erff(h * 0.70710678118654752440f));
        s += g * w2[n];
    }

    // Both 16-lane halves hold identical partials; reduce within a 16-group.
#pragma unroll
    for (int off = 8; off >= 1; off >>= 1) {
        s += __shfl_xor(s, off, 32);
    }

    if (lane == 0) {
        out[sample] = s + l2_bias[task];
    }
}

extern "C" void kernel_launch(void* const* d_in, const int* in_sizes, int n_in,
                              void* d_out, int out_size, void* d_ws, size_t ws_size,
                              hipStream_t stream) {
    const float* x        = (const float*)d_in[0];
    const int*   task_ids = (const int*)d_in[1];
    const float* l1_emb   = (const float*)d_in[2];
    const float* l1_bias  = (const float*)d_in[3];
    const float* l2_emb   = (const float*)d_in[4];
    const float* l2_bias  = (const float*)d_in[5];
    float* out = (float*)d_out;

    dim3 grid(BATCH / WAVES_PER_BLOCK);
    dim3 block(WAVES_PER_BLOCK * 32);
    mtl_fused_kernel<<<grid, block, 0, stream>>>(x, task_ids, l1_emb, l1_bias,
                                                 l2_emb, l2_bias, out);
}